// Criterion_85942295593390
// MI455X (gfx1250) — compile-verified
//
#include <hip/hip_runtime.h>
#include <stdint.h>

#define NUM_CLASSES 80
#define TOPK 10
#define B_ 16
#define SIDE 160
#define G_ 32
#define M_ (SIDE * SIDE)
#define REG_W 3.0f
#define INVALID_COST 1e9f

typedef float v2f __attribute__((ext_vector_type(2)));
typedef float v8f __attribute__((ext_vector_type(8)));

__device__ __forceinline__ float softplusf(float x) {
    // stable: max(x,0) + log(1 + exp(-|x|)); 1+t in (1,2] -> fast __logf is fine
    return fmaxf(x, 0.f) + __logf(1.f + __expf(-fabsf(x)));
}
__device__ __forceinline__ float sigmoidf(float x) {
    return 1.f / (1.f + __expf(-x));
}
__device__ __forceinline__ unsigned orderF(float f) {
    unsigned u = __float_as_uint(f);
    return (u & 0x80000000u) ? ~u : (u | 0x80000000u);
}

// Exact 32-lane sum using V_WMMA_F32_16X16X4_F32 with a ones B-matrix.
// A is 16x4: lane l<16 holds A[l][0]=v_l (K1=0); lane l>=16 holds A[l-16][2]=v_l (K3=0).
// D[r][j] = v_r + v_{r+16}. D VGPR r: lanes 0-15 -> row r, lanes 16-31 -> row r+8.
// Sum d[0..7] gives rows 0-7 (lanes<16) / rows 8-15 (lanes>=16); xor-16 shuffle completes it.
// Requires EXEC all-ones (call only at fully-converged points, full 256-thread blocks).
__device__ __forceinline__ float wave_reduce_sum_wmma(float v) {
    v2f a; a[0] = v;   a[1] = 0.f;
    v2f b; b[0] = 1.f; b[1] = 1.f;
    v8f c = {};
    v8f d = __builtin_amdgcn_wmma_f32_16x16x4_f32(false, a, false, b, (short)0, c, false, false);
    float s = ((d[0] + d[1]) + (d[2] + d[3])) + ((d[4] + d[5]) + (d[6] + d[7]));
    s += __shfl_xor(s, 16, 32);
    return s;
}

// ---------------- Kernel 0: init workspace ----------------
__global__ void k_init(unsigned long long* __restrict__ assign,
                       float* __restrict__ accum, int n) {
    int i = blockIdx.x * blockDim.x + threadIdx.x;
    if (i < n) assign[i] = ~0ull;
    if (i < 4) accum[i] = 0.f;
}

// ---------------- Kernel 1: valid_anchor[b,m] = any_g(anchor strictly inside gt) ----
__global__ void __launch_bounds__(256) k_valid(const float* __restrict__ gt_boxes,
                                               unsigned char* __restrict__ valid) {
    __shared__ float sgt[G_ * 4];
    int i = blockIdx.x * 256 + threadIdx.x;   // M_ % 256 == 0 -> block within one image
    int b = i / M_, m = i % M_;
    if (threadIdx.x < G_ * 4) sgt[threadIdx.x] = gt_boxes[b * G_ * 4 + threadIdx.x];
    __syncthreads();
    float ax = ((float)(m % SIDE) + 0.5f) * 8.f;
    float ay = ((float)(m / SIDE) + 0.5f) * 8.f;
    bool any = false;
    #pragma unroll 4
    for (int g = 0; g < G_; ++g) {
        float x1 = sgt[g*4+0], y1 = sgt[g*4+1], x2 = sgt[g*4+2], y2 = sgt[g*4+3];
        any = any || ((ax - x1 > 0.f) && (ay - y1 > 0.f) && (x2 - ax > 0.f) && (y2 - ay > 0.f));
    }
    valid[i] = any ? 1 : 0;
}

// ---------------- Kernel 2: SimOTA matcher, one block per (b,g) ----------------
__global__ void __launch_bounds__(256) k_match(
        const float* __restrict__ pred_cls, const float* __restrict__ pred_box,
        const float* __restrict__ gt_boxes, const int* __restrict__ gt_labels,
        const unsigned char* __restrict__ valid,
        unsigned long long* __restrict__ assign) {
    const int b = blockIdx.x / G_;
    const int g = blockIdx.x % G_;
    const int tid = threadIdx.x;

    __shared__ float sIou[2][256 * TOPK];
    __shared__ float sCost[2][256 * TOPK];
    __shared__ int   sIdx[2][256 * TOPK];

    const float gx1 = gt_boxes[(b*G_+g)*4+0];
    const float gy1 = gt_boxes[(b*G_+g)*4+1];
    const float gx2 = gt_boxes[(b*G_+g)*4+2];
    const float gy2 = gt_boxes[(b*G_+g)*4+3];
    const float garea = (gx2 - gx1) * (gy2 - gy1);
    const int label = gt_labels[b*G_ + g];

    float iouT[TOPK], costT[TOPK];
    int idxT[TOPK];
    #pragma unroll
    for (int r = 0; r < TOPK; ++r) { iouT[r] = -1.f; costT[r] = 3.4e38f; idxT[r] = 0; }

    const float* pb = pred_box + (size_t)b * M_ * 4;
    const float* pc = pred_cls + (size_t)b * M_ * NUM_CLASSES + label;
    const unsigned char* va = valid + (size_t)b * M_;

    for (int m = tid; m < M_; m += 256) {
        float4 bx = *(const float4*)(pb + (size_t)m * 4);
        __builtin_prefetch(pb + (size_t)(m + 256) * 4, 0, 0);
        float tlx = fmaxf(gx1, bx.x), tly = fmaxf(gy1, bx.y);
        float brx = fminf(gx2, bx.z), bry = fminf(gy2, bx.w);
        float iw = fmaxf(brx - tlx, 0.f), ih = fmaxf(bry - tly, 0.f);
        float inter = iw * ih;
        float a2 = (bx.z - bx.x) * (bx.w - bx.y);
        float iou = inter / fmaxf(garea + a2 - inter, 1e-8f);

        // top-10 IoUs (descending bubble insert, static indices -> VGPRs)
        float v = iou;
        #pragma unroll
        for (int r = 0; r < TOPK; ++r)
            if (v > iouT[r]) { float t = iouT[r]; iouT[r] = v; v = t; }

        float x   = pc[(size_t)m * NUM_CLASSES];
        float sp  = softplusf(x);        // softplus(x)
        float spn = sp - x;              // softplus(-x)
        float sg  = sigmoidf(x);
        float bce = iou * spn + (1.f - iou) * sp;
        float dfe = iou - sg;
        float cost = bce * dfe * dfe + REG_W * (-__logf(iou + 1e-8f));
        cost = va[m] ? cost : INVALID_COST;

        // top-10 lowest costs with anchor index
        float cv = cost; int ci = m;
        #pragma unroll
        for (int r = 0; r < TOPK; ++r)
            if (cv < costT[r]) {
                float t = costT[r]; int ti = idxT[r];
                costT[r] = cv; idxT[r] = ci; cv = t; ci = ti;
            }
    }

    #pragma unroll
    for (int r = 0; r < TOPK; ++r) {
        sIou[0][tid*TOPK+r]  = iouT[r];
        sCost[0][tid*TOPK+r] = costT[r];
        sIdx[0][tid*TOPK+r]  = idxT[r];
    }
    __syncthreads();

    // log2(256) ping-pong merge of sorted 10-lists
    int cur = 0;
    for (int half = 128; half >= 1; half >>= 1) {
        int nxt = cur ^ 1;
        if (tid < half) {
            {   // IoU lists: descending
                int ia = 0, ib = 0;
                #pragma unroll
                for (int k = 0; k < TOPK; ++k) {
                    float av = sIou[cur][tid*TOPK + ia];
                    float bv = sIou[cur][(tid+half)*TOPK + ib];
                    bool ta = av >= bv;
                    sIou[nxt][tid*TOPK + k] = ta ? av : bv;
                    ia += ta ? 1 : 0; ib += ta ? 0 : 1;
                }
            }
            {   // cost lists: ascending, carry index
                int ia = 0, ib = 0;
                #pragma unroll
                for (int k = 0; k < TOPK; ++k) {
                    float av = sCost[cur][tid*TOPK + ia];
                    float bv = sCost[cur][(tid+half)*TOPK + ib];
                    bool ta = av <= bv;
                    sCost[nxt][tid*TOPK + k] = ta ? av : bv;
                    sIdx[nxt][tid*TOPK + k] =
                        ta ? sIdx[cur][tid*TOPK + ia] : sIdx[cur][(tid+half)*TOPK + ib];
                    ia += ta ? 1 : 0; ib += ta ? 0 : 1;
                }
            }
        }
        __syncthreads();
        cur = nxt;
    }

    if (tid == 0) {
        float siou = 0.f;
        #pragma unroll
        for (int r = 0; r < TOPK; ++r) siou += sIou[cur][r];
        int dynk = (int)siou;                 // trunc, effective clip to [1, TOPK]
        if (dynk < 1) dynk = 1;
        if (dynk > TOPK) dynk = TOPK;
        for (int r = 0; r < dynk; ++r) {
            float c = sCost[cur][r];
            int   a = sIdx[cur][r];
            // key = (orderable cost)<<32 | g : atomicMin == argmin cost, lowest-g tie-break
            unsigned long long key = ((unsigned long long)orderF(c) << 32) | (unsigned)g;
            atomicMin(&assign[(size_t)b * M_ + a], key);
        }
    }
}

// ---------------- Kernel 3: focal + GIoU loss over all anchors ----------------
__global__ void __launch_bounds__(256) k_loss(
        const float* __restrict__ pred_cls, const float* __restrict__ pred_box,
        const float* __restrict__ gt_boxes, const int* __restrict__ gt_labels,
        const unsigned char* __restrict__ mask,
        const unsigned long long* __restrict__ assign,
        float* __restrict__ accum) {
    int i = blockIdx.x * 256 + threadIdx.x;
    int b = i / M_;
    unsigned long long key = assign[i];
    bool fg = (key != ~0ull);
    int g = (int)(key & 0xffffffffu);
    int label = fg ? gt_labels[b*G_ + g] : -1;

    const float* row = pred_cls + (size_t)i * NUM_CLASSES;
    __builtin_prefetch(row, 0, 0);
    __builtin_prefetch(row + 32, 0, 0);
    __builtin_prefetch(row + 64, 0, 0);

    // 4 independent accumulators -> breaks the serial v_add_f32 chain in the hot loop
    float acc0 = 0.f, acc1 = 0.f, acc2 = 0.f, acc3 = 0.f;
    #pragma unroll
    for (int q = 0; q < NUM_CLASSES / 4; ++q) {
        float4 xv = *(const float4*)(row + q * 4);
        float xs[4] = {xv.x, xv.y, xv.z, xv.w};
        float ts[4];
        #pragma unroll
        for (int j = 0; j < 4; ++j) {
            float x  = xs[j];
            float sp = softplusf(x);
            float p  = sigmoidf(x);
            int cls  = q * 4 + j;
            if (cls == label) {             // t=1: 0.25*(1-p)^2*softplus(-x)
                float q1 = 1.f - p;
                ts[j] = 0.25f * q1 * q1 * (sp - x);
            } else {                        // t=0: 0.75*p^2*softplus(x)
                ts[j] = 0.75f * p * p * sp;
            }
        }
        acc0 += ts[0]; acc1 += ts[1]; acc2 += ts[2]; acc3 += ts[3];
    }
    float clsSum = (acc0 + acc1) + (acc2 + acc3);
    if (mask[i]) clsSum = 0.f;

    float giouSum = 0.f, posCnt = 0.f;
    if (fg) {
        posCnt = 1.f;
        float4 pbv = *(const float4*)(pred_box + (size_t)i * 4);
        const float* gt = gt_boxes + ((size_t)b * G_ + g) * 4;
        float gx1 = gt[0], gy1 = gt[1], gx2 = gt[2], gy2 = gt[3];
        float tlx = fmaxf(pbv.x, gx1), tly = fmaxf(pbv.y, gy1);
        float brx = fminf(pbv.z, gx2), bry = fminf(pbv.w, gy2);
        float inter = fmaxf(brx - tlx, 0.f) * fmaxf(bry - tly, 0.f);
        float a1 = fmaxf(pbv.z - pbv.x, 0.f) * fmaxf(pbv.w - pbv.y, 0.f);
        float a2 = fmaxf(gx2 - gx1, 0.f) * fmaxf(gy2 - gy1, 0.f);
        float uni = a1 + a2 - inter;
        float iou = inter / fmaxf(uni, 1e-7f);
        float cw = fmaxf(pbv.z, gx2) - fminf(pbv.x, gx1);
        float ch = fmaxf(pbv.w, gy2) - fminf(pbv.y, gy1);
        float carea = fmaxf(fmaxf(cw, 0.f) * fmaxf(ch, 0.f), 1e-7f);
        float giou = iou - (carea - uni) / carea;
        giouSum = 1.f - giou;
    }

    // fully converged here: EXEC all-ones -> WMMA wave reductions are legal
    float wc = wave_reduce_sum_wmma(clsSum);
    float wg = wave_reduce_sum_wmma(giouSum);
    float wp = wave_reduce_sum_wmma(posCnt);

    __shared__ float sr[3][8];
    int lane = threadIdx.x & 31, wid = threadIdx.x >> 5;
    if (lane == 0) { sr[0][wid] = wc; sr[1][wid] = wg; sr[2][wid] = wp; }
    __syncthreads();
    if (threadIdx.x == 0) {
        float a0 = 0.f, a1 = 0.f, a2 = 0.f;
        #pragma unroll
        for (int w = 0; w < 8; ++w) { a0 += sr[0][w]; a1 += sr[1][w]; a2 += sr[2][w]; }
        atomicAdd(&accum[0], a0);
        atomicAdd(&accum[1], a1);
        atomicAdd(&accum[2], a2);
    }
}

// ---------------- Kernel 4: finalize ----------------
__global__ void k_final(const float* __restrict__ accum, float* __restrict__ out) {
    float pos = fmaxf(accum[2], 1.f);
    out[0] = accum[0] / pos;
    out[1] = accum[1] / pos;
}

extern "C" void kernel_launch(void* const* d_in, const int* in_sizes, int n_in,
                              void* d_out, int out_size, void* d_ws, size_t ws_size,
                              hipStream_t stream) {
    const float* pred_cls        = (const float*)d_in[0];
    const float* pred_box        = (const float*)d_in[1];
    /* d_in[2] anchors: computed analytically ((idx+0.5)*stride), not read */
    const float* gt_boxes        = (const float*)d_in[3];
    const int*   gt_labels       = (const int*)d_in[4];
    const unsigned char* mask    = (const unsigned char*)d_in[5];
    float* out = (float*)d_out;

    char* ws = (char*)d_ws;
    unsigned long long* assign = (unsigned long long*)ws;                         // B*M u64
    float* accum               = (float*)(ws + (size_t)B_ * M_ * 8);              // 4 f32
    unsigned char* valid       = (unsigned char*)(ws + (size_t)B_ * M_ * 8 + 64); // B*M u8

    const int nBM = B_ * M_;   // 409600, divisible by 256
    k_init <<<(nBM + 255) / 256, 256, 0, stream>>>(assign, accum, nBM);
    k_valid<<<nBM / 256,        256, 0, stream>>>(gt_boxes, valid);
    k_match<<<B_ * G_,          256, 0, stream>>>(pred_cls, pred_box, gt_boxes,
                                                  gt_labels, valid, assign);
    k_loss <<<nBM / 256,        256, 0, stream>>>(pred_cls, pred_box, gt_boxes,
                                                  gt_labels, mask, assign, accum);
    k_final<<<1, 1, 0, stream>>>(accum, out);
}